// NodeModel_9440338116647
// MI455X (gfx1250) — compile-verified
//
#include <hip/hip_runtime.h>
#include <hip/hip_bf16.h>

typedef __attribute__((ext_vector_type(2))) float v2f;
typedef __attribute__((ext_vector_type(8))) float v8f;

#define EDGE_BLOCK 256           // 8 waves/block
#define WAVES_PB   8

// ---------------------------------------------------------------------------
// Zero workspace (accumulator + counts) — must run every launch (graph replay)
// ---------------------------------------------------------------------------
__global__ void zero_kernel(float* __restrict__ p, long long n) {
    long long i = (long long)blockIdx.x * blockDim.x + threadIdx.x;
    long long stride = (long long)gridDim.x * blockDim.x;
    for (; i < n; i += stride) p[i] = 0.0f;
}

// ---------------------------------------------------------------------------
// Edge kernel: per wave = 32 edges = two 16-edge WMMA tiles.
//   layer1: h = relu(x[row] @ W1a + b1a)   via V_WMMA_F32_16X16X4_F32 (K=3 pad 4)
//   layer2: h = h @ W1b + b1b              via 4 chained K=4 f32 WMMAs
//   scatter: atomic f32 adds into s[col][16], count into cnt[col]
// All lane-layout shuffling goes through per-wave LDS buffers, branch-free,
// so EXEC is all-1s across every WMMA.
// ---------------------------------------------------------------------------
__global__ __launch_bounds__(EDGE_BLOCK) void edge_kernel(
    const float* __restrict__ x,
    const int*   __restrict__ ei,        // [2, E] (row-major: row then col)
    const float* __restrict__ W1a, const float* __restrict__ b1a,
    const float* __restrict__ W1b, const float* __restrict__ b1b,
    float* __restrict__ sacc, float* __restrict__ cnt,
    long long E)
{
    __shared__ float w1a[4 * 16];                 // [K=4 padded][N=16]
    __shared__ float w1b[16 * 16];                // [K=16][N=16]
    __shared__ float bias1a[16], bias1b[16];
    __shared__ float xbuf[WAVES_PB][32][4];       // gathered x, padded K
    __shared__ float hbuf[WAVES_PB][32][16];      // layer1 output, row-major
    __shared__ int   colbuf[WAVES_PB][32];

    const int tid  = threadIdx.x;
    const int wave = tid >> 5;
    const int lane = tid & 31;

    // Stage weights (block-shared)
    if (tid < 48)                 w1a[tid] = W1a[tid];
    else if (tid < 64)            w1a[tid] = 0.0f;        // pad K=3 row
    if (tid < 256)                w1b[tid] = W1b[tid];
    if (tid < 16) { bias1a[tid] = b1a[tid]; bias1b[tid] = b1b[tid]; }

    // Gather edge endpoints + source features
    const long long e = (long long)blockIdx.x * EDGE_BLOCK + wave * 32 + lane;
    const int row = ei[e];
    const int col = ei[e + E];
    const float* xp = x + (long long)row * 3;
    xbuf[wave][lane][0] = xp[0];
    xbuf[wave][lane][1] = xp[1];
    xbuf[wave][lane][2] = xp[2];
    xbuf[wave][lane][3] = 0.0f;                   // K=3 pad
    colbuf[wave][lane]  = col;
    atomicAdd(&cnt[col], 1.0f);                   // degree count
    __syncthreads();

    const int sel = (lane >> 4) & 1;              // 0: lanes 0-15, 1: lanes 16-31
    const int nn  = lane & 15;

    // -------- layer 1: [16x4] x [4x16] per tile --------
    #pragma unroll
    for (int t = 0; t < 2; ++t) {
        const int me = t * 16 + nn;               // A: M over lanes
        v2f a1; a1.x = xbuf[wave][me][2 * sel];           // K0 / K2
                a1.y = xbuf[wave][me][2 * sel + 1];       // K1 / K3(=0)
        v2f b1; b1.x = w1a[(2 * sel) * 16 + nn];
                b1.y = w1a[(2 * sel + 1) * 16 + nn];
        v8f c0 = {};
        v8f d1 = __builtin_amdgcn_wmma_f32_16x16x4_f32(
                     false, a1, false, b1, (short)0, c0, false, false);
        const float bb = bias1a[nn];
        #pragma unroll
        for (int i = 0; i < 8; ++i) {
            float v = d1[i] + bb;
            v = v > 0.0f ? v : 0.0f;              // ReLU
            const int m = t * 16 + i + 8 * sel;   // D: M = vgpr + 8*(lane>=16)
            hbuf[wave][m][nn] = v;                // row-major restage
        }
    }
    __syncthreads();

    // -------- layer 2: [16x16] x [16x16] per tile, 4 chained K=4 WMMAs --------
    #pragma unroll
    for (int t = 0; t < 2; ++t) {
        v8f acc = {};
        #pragma unroll
        for (int s4 = 0; s4 < 4; ++s4) {
            const int kb = s4 * 4 + 2 * sel;
            v2f a2; a2.x = hbuf[wave][t * 16 + nn][kb];
                    a2.y = hbuf[wave][t * 16 + nn][kb + 1];
            v2f b2; b2.x = w1b[kb * 16 + nn];
                    b2.y = w1b[(kb + 1) * 16 + nn];
            acc = __builtin_amdgcn_wmma_f32_16x16x4_f32(
                      false, a2, false, b2, (short)0, acc, false, false);
        }
        const float bb = bias1b[nn];
        #pragma unroll
        for (int i = 0; i < 8; ++i) {
            const int m = t * 16 + i + 8 * sel;
            const int c = colbuf[wave][m];
            atomicAdd(&sacc[(long long)c * 16 + nn], acc[i] + bb);
        }
    }
}

// ---------------------------------------------------------------------------
// Scalar tail for edges beyond the last full 256-edge block (E=16M -> unused)
// ---------------------------------------------------------------------------
__global__ void edge_tail_kernel(
    const float* __restrict__ x, const int* __restrict__ ei,
    const float* __restrict__ W1a, const float* __restrict__ b1a,
    const float* __restrict__ W1b, const float* __restrict__ b1b,
    float* __restrict__ sacc, float* __restrict__ cnt,
    long long start, long long E)
{
    long long e = start + (long long)blockIdx.x * blockDim.x + threadIdx.x;
    if (e >= E) return;
    const int row = ei[e];
    const int col = ei[e + E];
    const float x0 = x[(long long)row * 3 + 0];
    const float x1 = x[(long long)row * 3 + 1];
    const float x2 = x[(long long)row * 3 + 2];
    float h1[16];
    for (int j = 0; j < 16; ++j) {
        float a = b1a[j] + x0 * W1a[0 * 16 + j] + x1 * W1a[1 * 16 + j] + x2 * W1a[2 * 16 + j];
        h1[j] = a > 0.0f ? a : 0.0f;
    }
    for (int j = 0; j < 16; ++j) {
        float a = b1b[j];
        for (int k = 0; k < 16; ++k) a += h1[k] * W1b[k * 16 + j];
        atomicAdd(&sacc[(long long)col * 16 + j], a);
    }
    atomicAdd(&cnt[col], 1.0f);
}

// ---------------------------------------------------------------------------
// Node kernel: agg = s/cnt (0 if cnt==0), z=[x,agg] -> 16 -> 16 -> 3
// Weights staged in LDS; ~0.6 GFLOP total, negligible.
// ---------------------------------------------------------------------------
__global__ __launch_bounds__(256) void node_kernel(
    const float* __restrict__ x,
    const float* __restrict__ sacc, const float* __restrict__ cnt,
    const float* __restrict__ W2a, const float* __restrict__ b2a,
    const float* __restrict__ W2b, const float* __restrict__ b2b,
    const float* __restrict__ W2c, const float* __restrict__ b2c,
    float* __restrict__ out, int n)
{
    __shared__ float wa[19 * 16], wb[16 * 16], wc[16 * 3];
    __shared__ float ba[16], bb[16], bc[3];
    const int tid = threadIdx.x;
    for (int i = tid; i < 19 * 16; i += 256) wa[i] = W2a[i];
    for (int i = tid; i < 16 * 16; i += 256) wb[i] = W2b[i];
    for (int i = tid; i < 16 * 3;  i += 256) wc[i] = W2c[i];
    if (tid < 16) { ba[tid] = b2a[tid]; bb[tid] = b2b[tid]; }
    if (tid < 3)  bc[tid] = b2c[tid];
    __syncthreads();

    const int i = blockIdx.x * 256 + tid;
    if (i >= n) return;

    float z[19];
    z[0] = x[(long long)i * 3 + 0];
    z[1] = x[(long long)i * 3 + 1];
    z[2] = x[(long long)i * 3 + 2];
    const float c   = cnt[i];
    const float inv = c > 0.0f ? 1.0f / c : 0.0f;    // mean (0 for empty segs)
    #pragma unroll
    for (int k = 0; k < 16; ++k) z[3 + k] = sacc[(long long)i * 16 + k] * inv;

    float h1[16];
    #pragma unroll
    for (int j = 0; j < 16; ++j) {
        float a = ba[j];
        #pragma unroll
        for (int k = 0; k < 19; ++k) a += z[k] * wa[k * 16 + j];
        h1[j] = a > 0.0f ? a : 0.0f;
    }
    float h2[16];
    #pragma unroll
    for (int j = 0; j < 16; ++j) {
        float a = bb[j];
        #pragma unroll
        for (int k = 0; k < 16; ++k) a += h1[k] * wb[k * 16 + j];
        h2[j] = a > 0.0f ? a : 0.0f;
    }
    #pragma unroll
    for (int j = 0; j < 3; ++j) {
        float a = bc[j];
        #pragma unroll
        for (int k = 0; k < 16; ++k) a += h2[k] * wc[k * 3 + j];
        out[(long long)i * 3 + j] = a;
    }
}

// ---------------------------------------------------------------------------
extern "C" void kernel_launch(void* const* d_in, const int* in_sizes, int n_in,
                              void* d_out, int out_size, void* d_ws, size_t ws_size,
                              hipStream_t stream) {
    const float* x   = (const float*)d_in[0];
    const int*   ei  = (const int*)  d_in[1];   // [2,E]
    // d_in[2]=edge_attr, d_in[3]=u, d_in[4]=batch : unused by the reference math
    const float* W1a = (const float*)d_in[5];
    const float* b1a = (const float*)d_in[6];
    const float* W1b = (const float*)d_in[7];
    const float* b1b = (const float*)d_in[8];
    const float* W2a = (const float*)d_in[9];
    const float* b2a = (const float*)d_in[10];
    const float* W2b = (const float*)d_in[11];
    const float* b2b = (const float*)d_in[12];
    const float* W2c = (const float*)d_in[13];
    const float* b2c = (const float*)d_in[14];

    const int       n = in_sizes[0] / 3;             // 500000 nodes
    const long long E = (long long)in_sizes[1] / 2;  // 16000000 edges

    float* sacc = (float*)d_ws;                      // [n,16] f32 accumulator
    float* cnt  = sacc + (long long)n * 16;          // [n]    degree counts

    // 1) zero accumulators (34 MB, L2-resident)
    zero_kernel<<<2048, 256, 0, stream>>>(sacc, (long long)n * 17);

    // 2) WMMA edge MLP + scatter
    const long long full_blocks = E / EDGE_BLOCK;
    if (full_blocks > 0)
        edge_kernel<<<(unsigned)full_blocks, EDGE_BLOCK, 0, stream>>>(
            x, ei, W1a, b1a, W1b, b1b, sacc, cnt, E);
    const long long tail_start = full_blocks * EDGE_BLOCK;
    const long long tail = E - tail_start;
    if (tail > 0)
        edge_tail_kernel<<<(unsigned)((tail + 255) / 256), 256, 0, stream>>>(
            x, ei, W1a, b1a, W1b, b1b, sacc, cnt, tail_start, E);

    // 3) node MLP
    node_kernel<<<(n + 255) / 256, 256, 0, stream>>>(
        x, sacc, cnt, W2a, b2a, W2b, b2b, W2c, b2c, (float*)d_out, n);
}